// SplitTableBatchedEmbeddingBagsCodegen_46153718563288
// MI455X (gfx1250) — compile-verified
//
#include <hip/hip_runtime.h>
#include <hip/hip_bf16.h>

// Embedding-bag SUM pooling (FBGEMM split-table forward), MI455X / gfx1250.
// One wave32 per bag; lane owns float4 -> 32 lanes cover D=128 exactly.
// Scalar (SGPR) index path, 4-deep b128 gather pipeline, full-coverage
// global_prefetch_b8 lookahead (distance 8), non-temporal output stores
// to preserve L2 residency of the 409.6MB table set (~1.25x row reuse).

namespace {

constexpr int kT = 8;        // tables
constexpr int kE = 100000;   // rows per table
constexpr int kD = 128;      // embedding dim
constexpr int kB = 4096;     // batch
constexpr int kBags = kT * kB;           // 32768 bags
constexpr int kWavesPerBlock = 8;        // 256 threads
constexpr int kThreads = kWavesPerBlock * 32;

typedef float float4v __attribute__((ext_vector_type(4)));

__device__ __forceinline__ float4v load4(const float* __restrict__ p) {
    return *(const float4v* __restrict__)p;   // -> global_load_b128
}

} // namespace

__global__ __launch_bounds__(kThreads)
void emb_bag_sum_kernel(const float* __restrict__ weights,
                        const int* __restrict__ indices,
                        const int* __restrict__ offsets,
                        float* __restrict__ out)
{
    const int lane = threadIdx.x & 31;
    // Force wave-uniformity so offsets/indices fetches become scalar loads.
    const int wave_in_blk = __builtin_amdgcn_readfirstlane((int)(threadIdx.x >> 5));
    const int bag = (int)blockIdx.x * kWavesPerBlock + wave_in_blk;
    if (bag >= kBags) return;

    const int t = bag >> 12;          // bag / B   (B = 4096)
    const int b = bag & (kB - 1);     // bag % B

    const int start = offsets[bag];
    const int end   = offsets[bag + 1];

    const float* __restrict__ tbl = weights + (size_t)t * ((size_t)kE * kD);
    const int col = lane << 2;        // 4 floats per lane (coalesced 512B/row)

    float4v a0 = {0.f, 0.f, 0.f, 0.f};
    float4v a1 = a0, a2 = a0, a3 = a0;

    int i = start;

    // Main pipeline: 4 independent 128-bit gathers in flight per wave.
    // Prefetch ALL rows of the iteration 2 steps ahead (distance 8) into
    // cache (locality 3 -> keep in L2 where the ~1.25x row reuse lives).
    // global_prefetch_b8 does not touch LOADcnt: free latency hiding.
    for (; i + 4 <= end; i += 4) {
        const int r0 = indices[i + 0];
        const int r1 = indices[i + 1];
        const int r2 = indices[i + 2];
        const int r3 = indices[i + 3];

        if (i + 12 <= end) {
            const int p0 = indices[i + 8];
            const int p1 = indices[i + 9];
            const int p2 = indices[i + 10];
            const int p3 = indices[i + 11];
            __builtin_prefetch(tbl + (size_t)p0 * kD + col, 0, 3);
            __builtin_prefetch(tbl + (size_t)p1 * kD + col, 0, 3);
            __builtin_prefetch(tbl + (size_t)p2 * kD + col, 0, 3);
            __builtin_prefetch(tbl + (size_t)p3 * kD + col, 0, 3);
        }

        const float4v v0 = load4(tbl + (size_t)r0 * kD + col);
        const float4v v1 = load4(tbl + (size_t)r1 * kD + col);
        const float4v v2 = load4(tbl + (size_t)r2 * kD + col);
        const float4v v3 = load4(tbl + (size_t)r3 * kD + col);

        a0 += v0; a1 += v1; a2 += v2; a3 += v3;
    }
    // Remainder (0..3 rows).
    for (; i < end; ++i) {
        const int r = indices[i];
        a0 += load4(tbl + (size_t)r * kD + col);
    }

    const float4v acc = (a0 + a1) + (a2 + a3);

    // Output layout: [B, T*D]; each (t,b) segment written exactly once.
    // Non-temporal store: output is write-once, never re-read by this kernel;
    // keep L2 capacity for embedding rows instead.
    float4v* __restrict__ o =
        (float4v*)(out + (size_t)b * (kT * kD) + (size_t)t * kD + col);
    __builtin_nontemporal_store(acc, o);
}

extern "C" void kernel_launch(void* const* d_in, const int* in_sizes, int n_in,
                              void* d_out, int out_size, void* d_ws, size_t ws_size,
                              hipStream_t stream) {
    (void)in_sizes; (void)n_in; (void)out_size; (void)d_ws; (void)ws_size;
    const float* weights = (const float*)d_in[0];  // [T, E, D] fp32
    const int*   indices = (const int*)d_in[1];    // [N] int32
    const int*   offsets = (const int*)d_in[2];    // [T*B+1] int32
    float* out = (float*)d_out;                    // [B, T*D] fp32

    const int blocks = kBags / kWavesPerBlock;     // 4096 blocks x 256 threads
    emb_bag_sum_kernel<<<blocks, kThreads, 0, stream>>>(weights, indices, offsets, out);
}